// MASKLoss_10187662426678
// MI455X (gfx1250) — compile-verified
//
#include <hip/hip_runtime.h>

#define ALPHA_F    0.25f
#define CLAMP_F    1.0e-4f
#define NBLK       1024
#define TPB        256

typedef __attribute__((ext_vector_type(2))) float v2f;
typedef __attribute__((ext_vector_type(8))) float v8f;

// ---------------------------------------------------------------------------
// Deterministic wave32 sum via V_WMMA_F32_16X16X4_F32.
// A (16x4 f32): lanes 0-15 hold K=0 (vgpr0), K=1 (vgpr1); lanes 16-31 hold K=2,3.
// Lane partial in vgpr0 slot, zero in vgpr1; B = ones => D[m,n] = s_m + s_{m+16}.
// Sum the 8 D vgprs per lane (half-sums), then xor-16 to get the full wave sum.
// EXEC is all ones here (called at uniform end of kernel). Fixed reduction order.
// ---------------------------------------------------------------------------
__device__ __forceinline__ float wave_sum32(float x) {
  v2f A; A.x = x;    A.y = 0.0f;
  v2f B; B.x = 1.0f; B.y = 1.0f;
  v8f C = {};
  v8f D = __builtin_amdgcn_wmma_f32_16x16x4_f32(false, A, false, B,
                                                (short)0, C, false, false);
  float h = D[0] + D[1] + D[2] + D[3] + D[4] + D[5] + D[6] + D[7];
  h += __shfl_xor(h, 16, 32);
  return h;
}

__device__ __forceinline__ float clamped_sigmoid(float x) {
  float p = 1.0f / (1.0f + expf(-x));
  return fminf(fmaxf(p, CLAMP_F), 1.0f - CLAMP_F);
}

// b = log(1-p0)*p0^2 term for a negative (maskless) row
__device__ __forceinline__ float neg_term(float x) {
  float p0 = clamped_sigmoid(x);
  return logf(1.0f - p0) * p0 * p0;
}

// s^c with the reference's pow edge semantics we care about (c==0 -> 1).
__device__ __forceinline__ float pow_sc(float ls /*log2(s)*/, float c) {
  return (c == 0.0f) ? 1.0f : exp2f(c * ls);
}

// ---------------------------------------------------------------------------
// Pass 1: stream the 128MB mask once with B128 per-lane loads (4 rows / wave /
// iteration). Rebuild each row's 32-bit mask with 3 xor-OR shuffles inside the
// aligned 8-lane row group. Accumulate conf[g] maxima, has bits, neg-row loss,
// and store the compressed bitmask for passes 2/3.
// ---------------------------------------------------------------------------
__global__ void pass1_kernel(const float* __restrict__ logits,
                             const float* __restrict__ scores,
                             const int*   __restrict__ mask,
                             unsigned*    __restrict__ bmask,
                             int use_bm,
                             float*       __restrict__ conf,      // [32] raw max (uint-bits cmp, >=0)
                             unsigned*    __restrict__ hasmask,   // [1]
                             double*      __restrict__ negPart,   // [NBLK]
                             int N) {
  const int lane   = threadIdx.x & 31;
  const int warp   = threadIdx.x >> 5;
  const int wid    = (blockIdx.x * blockDim.x + threadIdx.x) >> 5;
  const int nwaves = (gridDim.x * blockDim.x) >> 5;

  const int rgrp    = lane >> 3;        // row within 4-row chunk handled by this lane
  const int colBase = (lane & 7) * 4;   // first of the 4 columns this lane covers

  float    cmax0 = 0.0f, cmax1 = 0.0f, cmax2 = 0.0f, cmax3 = 0.0f;
  unsigned hasbits = 0u;
  float    negaccF = 0.0f;

  const int4* __restrict__ m4 = (const int4*)mask;
  const int nChunks = N >> 2;           // 4 rows per chunk

  for (int c = wid; c < nChunks; c += nwaves) {
    // one B128 per lane: wave covers rows 4c..4c+3 (512 bytes)
    if (c + nwaves < nChunks)
      __builtin_prefetch(&m4[(size_t)(c + nwaves) * 32u + lane], 0, 1);
    int4 v = m4[(size_t)c * 32u + lane];

    unsigned nib = (v.x > 0 ? 1u : 0u) | (v.y > 0 ? 2u : 0u) |
                   (v.z > 0 ? 4u : 0u) | (v.w > 0 ? 8u : 0u);
    unsigned bm = nib << colBase;
    bm |= __shfl_xor(bm, 1, 32);
    bm |= __shfl_xor(bm, 2, 32);
    bm |= __shfl_xor(bm, 4, 32);        // full row mask in every lane of the group
    hasbits |= bm;

    const int row = c * 4 + rgrp;
    float s = scores[row];
    if (nib & 1u) cmax0 = fmaxf(cmax0, s);
    if (nib & 2u) cmax1 = fmaxf(cmax1, s);
    if (nib & 4u) cmax2 = fmaxf(cmax2, s);
    if (nib & 8u) cmax3 = fmaxf(cmax3, s);

    if ((lane & 7) == 0) {
      if (use_bm) bmask[row] = bm;      // lanes 0,8,16,24 -> 4 consecutive dwords
      if (bm == 0u) negaccF += neg_term(logits[row]);
    }
  }

  // scalar tail for N % 4 rows (lane == column mapping)
  float cmaxT = 0.0f;
  {
    const int N4 = N & ~3;
    const int i  = N4 + wid;
    if (i < N) {
      int m = mask[(size_t)i * 32u + lane] > 0;
      unsigned bm = __builtin_amdgcn_ballot_w32(m != 0);
      hasbits |= bm;
      if (m) cmaxT = scores[i];
      if (lane == 0) {
        if (use_bm) bmask[i] = bm;
        if (bm == 0u) negaccF += neg_term(logits[i]);
      }
    }
  }

  // order-independent cross-wave combines (exact): bitwise max on non-negative f32
  atomicMax((unsigned*)&conf[colBase + 0], __float_as_uint(cmax0));
  atomicMax((unsigned*)&conf[colBase + 1], __float_as_uint(cmax1));
  atomicMax((unsigned*)&conf[colBase + 2], __float_as_uint(cmax2));
  atomicMax((unsigned*)&conf[colBase + 3], __float_as_uint(cmax3));
  atomicMax((unsigned*)&conf[lane],        __float_as_uint(cmaxT));
  hasbits |= __shfl_xor(hasbits, 8, 32);
  hasbits |= __shfl_xor(hasbits, 16, 32);
  if (lane == 0) atomicOr(hasmask, hasbits);

  // deterministic wave sum of neg-loss partials on the matrix pipe
  float wneg = wave_sum32(negaccF);

  __shared__ double red[TPB / 32];
  if (lane == 0) red[warp] = (double)wneg;
  __syncthreads();
  if (threadIdx.x == 0) {
    double t = 0.0;
    for (int k = 0; k < TPB / 32; ++k) t += red[k];
    negPart[blockIdx.x] = t;
  }
}

// conf/vmax finalize: columns with no mask become 1.0
__global__ void fin_cv_kernel(float* __restrict__ arr,
                              const unsigned* __restrict__ hasmask) {
  int g = threadIdx.x;
  if (g < 32 && !((*hasmask >> g) & 1u)) arr[g] = 1.0f;
}

// ---------------------------------------------------------------------------
// Pass 2: vmax[g] = max over masked rows of scores[i]^conf[g] * IoU[i]
// ---------------------------------------------------------------------------
__global__ void pass2_kernel(const float* __restrict__ scores,
                             const float* __restrict__ iou,
                             const int*   __restrict__ mask,
                             const unsigned* __restrict__ bmask,
                             int use_bm,
                             const float* __restrict__ conf,
                             float*       __restrict__ vmax,
                             int N) {
  const int lane   = threadIdx.x & 31;
  const int wid    = (blockIdx.x * blockDim.x + threadIdx.x) >> 5;
  const int nwaves = (gridDim.x * blockDim.x) >> 5;

  const float cl = conf[lane];
  float vm = 0.0f;  // val >= 0 always

  for (int i = wid; i < N; i += nwaves) {
    unsigned bm = use_bm ? bmask[i]
                         : __builtin_amdgcn_ballot_w32(mask[(size_t)i * 32u + lane] > 0);
    if (bm == 0u) continue;
    float s  = scores[i];
    float u  = iou[i];
    float ls = log2f(s);
    if ((bm >> lane) & 1u) {
      float val = pow_sc(ls, cl) * u;
      vm = fmaxf(vm, val);
    }
  }
  atomicMax((unsigned*)&vmax[lane], __float_as_uint(vm));
}

// ---------------------------------------------------------------------------
// Pass 3: S = sum over masked (i,g) of  a_i*w^2 + b_i*(1-w)^2
//   a = log(p0)*(1-p0)^2 , b = log(1-p0)*p0^2 , w = clip((val+C)/(vmax_g+C))
// ---------------------------------------------------------------------------
__global__ void pass3_kernel(const float* __restrict__ logits,
                             const float* __restrict__ scores,
                             const float* __restrict__ iou,
                             const int*   __restrict__ mask,
                             const unsigned* __restrict__ bmask,
                             int use_bm,
                             const float* __restrict__ conf,
                             const float* __restrict__ vmaxA,
                             double*      __restrict__ sPart,   // [NBLK]
                             int N) {
  const int lane   = threadIdx.x & 31;
  const int warp   = threadIdx.x >> 5;
  const int wid    = (blockIdx.x * blockDim.x + threadIdx.x) >> 5;
  const int nwaves = (gridDim.x * blockDim.x) >> 5;

  const float cl   = conf[lane];
  const float invv = 1.0f / (vmaxA[lane] + CLAMP_F);
  const float wlo  = CLAMP_F, whi = 1.0f - CLAMP_F;

  float acc = 0.0f;   // <= ~N/nwaves (~122) terms per lane: f32 is plenty

  for (int i = wid; i < N; i += nwaves) {
    unsigned bm = use_bm ? bmask[i]
                         : __builtin_amdgcn_ballot_w32(mask[(size_t)i * 32u + lane] > 0);
    if (bm == 0u) continue;
    float s  = scores[i];
    float u  = iou[i];
    float p0 = clamped_sigmoid(logits[i]);
    float omp = 1.0f - p0;
    float a  = logf(p0) * omp * omp;
    float b  = logf(omp) * p0 * p0;
    float ls = log2f(s);
    if ((bm >> lane) & 1u) {
      float val = pow_sc(ls, cl) * u;
      float w   = fminf(fmaxf((val + CLAMP_F) * invv, wlo), whi);
      float ow  = 1.0f - w;
      acc += a * w * w + b * ow * ow;
    }
  }

  // deterministic wave reduction on the matrix pipe, then fixed-order block sum
  float wsum = wave_sum32(acc);

  __shared__ double red[TPB / 32];
  if (lane == 0) red[warp] = (double)wsum;
  __syncthreads();
  if (threadIdx.x == 0) {
    double t = 0.0;
    for (int k = 0; k < TPB / 32; ++k) t += red[k];
    sPart[blockIdx.x] = t;
  }
}

// ---------------------------------------------------------------------------
// Init / finalize
// ---------------------------------------------------------------------------
__global__ void init_kernel(float* conf, float* vmax, unsigned* hasmask,
                            double* negPart, double* sPart, int nblk) {
  int t = blockIdx.x * blockDim.x + threadIdx.x;
  int stride = gridDim.x * blockDim.x;
  if (t < 32) { conf[t] = 0.0f; vmax[t] = 0.0f; }
  if (t == 32) *hasmask = 0u;
  for (int i = t; i < nblk; i += stride) { negPart[i] = 0.0; sPart[i] = 0.0; }
}

__global__ void final_kernel(const double* __restrict__ negPart,
                             const double* __restrict__ sPart,
                             int nblk,
                             const int* __restrict__ nposPtr,
                             float* __restrict__ out) {
  if (blockIdx.x == 0 && threadIdx.x == 0) {
    double ns = 0.0, s = 0.0;
    for (int i = 0; i < nblk; ++i) ns += negPart[i];   // fixed order
    for (int i = 0; i < nblk; ++i) s  += sPart[i];
    double npos = (double)nposPtr[0];
    // neg_loss = -0.75*ns ; pos+box = -0.25*s
    out[0] = (float)((-0.75 * ns - 0.25 * s) / npos);
  }
}

// ---------------------------------------------------------------------------
// Launch
// ---------------------------------------------------------------------------
extern "C" void kernel_launch(void* const* d_in, const int* in_sizes, int n_in,
                              void* d_out, int out_size, void* d_ws, size_t ws_size,
                              hipStream_t stream) {
  const float* logits = (const float*)d_in[0];
  const float* scores = (const float*)d_in[1];
  const float* ioum   = (const float*)d_in[2];
  const int*   mask   = (const int*)d_in[3];
  // d_in[4] = gt_labels: all zeros and scores has one column -> index 0 always.
  const int*   npos   = (const int*)d_in[5];
  float*       out    = (float*)d_out;
  const int N = in_sizes[0];

  // workspace layout
  char* w = (char*)d_ws;
  float*    conf    = (float*)(w + 0);          // 32 f32
  float*    vmax    = (float*)(w + 128);        // 32 f32
  unsigned* hasmask = (unsigned*)(w + 256);     // 1 u32 (pad to 320)
  double*   negPart = (double*)(w + 320);       // NBLK f64
  double*   sPart   = negPart + NBLK;           // NBLK f64
  unsigned* bmask   = (unsigned*)(w + 320 + 2 * NBLK * sizeof(double));
  const size_t base_need = 320 + 2 * (size_t)NBLK * sizeof(double);
  const int use_bm = (ws_size >= base_need + 4ull * (size_t)N) ? 1 : 0;

  init_kernel<<<8, TPB, 0, stream>>>(conf, vmax, hasmask, negPart, sPart, NBLK);

  pass1_kernel<<<NBLK, TPB, 0, stream>>>(logits, scores, mask,
                                         use_bm ? bmask : (unsigned*)nullptr, use_bm,
                                         conf, hasmask, negPart, N);
  fin_cv_kernel<<<1, 32, 0, stream>>>(conf, hasmask);

  pass2_kernel<<<NBLK, TPB, 0, stream>>>(scores, ioum, mask, bmask, use_bm,
                                         conf, vmax, N);
  fin_cv_kernel<<<1, 32, 0, stream>>>(vmax, hasmask);

  pass3_kernel<<<NBLK, TPB, 0, stream>>>(logits, scores, ioum, mask, bmask, use_bm,
                                         conf, vmax, sPart, N);

  final_kernel<<<1, 32, 0, stream>>>(negPart, sPart, NBLK, npos, out);
}